// SpectralFilter_11948599017837
// MI455X (gfx1250) — compile-verified
//
#include <hip/hip_runtime.h>
#include <math.h>

// CDNA5 / gfx1250, wave32. fp32 matrix path via V_WMMA_F32_16X16X4_F32.
typedef float v2f __attribute__((ext_vector_type(2)));
typedef float v8f __attribute__((ext_vector_type(8)));

#define BM 128   // workgroup rows   (4 waves x 32 rows)
#define BN 64    // workgroup cols   (4 x 16 accumulator columns)
#define BK 16    // K chunk
#define ALD 20   // LDS row stride for A tile (floats): 16B-aligned, conflict-free
#define NPAIR 8  // BK/2 k-pairs in B tile
#define BPLD 160 // LDS pair-row stride for B tile: 160 % 64 == 32 -> halves hit
                 // disjoint bank sets, b64 frag loads are conflict-free

// ---------------- staging helpers (into one LDS buffer) ---------------------
__device__ __forceinline__
void stage_A_n(float (*As)[ALD], const float* __restrict__ Ab,
               int row0, int lda, int kc, int tid)
{
  // A[row][k] row-major -> As[row][k]; 128x16 floats = 512 float4
  #pragma unroll
  for (int it = 0; it < 4; ++it) {
    int q = tid + it * 128;
    int r = q >> 2, k4 = (q & 3) << 2;
    float4 v = *(const float4*)(Ab + (long long)(row0 + r) * lda + kc + k4);
    *(float4*)&As[r][k4] = v;
  }
}

__device__ __forceinline__
void stage_A_t(float (*As)[ALD], const float* __restrict__ Ab,
               int row0, int lda, int kc, int tid)
{
  // memory At[k][m] (k-row stride lda) -> transpose into As[m][k]
  #pragma unroll
  for (int it = 0; it < 4; ++it) {
    int q = tid + it * 128;
    int k = q >> 5, m4 = (q & 31) << 2;
    float4 v = *(const float4*)(Ab + (long long)(kc + k) * lda + row0 + m4);
    As[m4 + 0][k] = v.x; As[m4 + 1][k] = v.y;
    As[m4 + 2][k] = v.z; As[m4 + 3][k] = v.w;
  }
}

__device__ __forceinline__
void stage_B_pairs(float (*Bp)[BPLD], const float* __restrict__ Bb,
                   int col0, int ldb, int kc, int tid)
{
  // B[k][n] row-major -> k-pair interleaved Bp[k>>1][2*col + (k&1)]
  // so a B fragment is ONE aligned ds_load_b64 per lane (no packing movs).
  #pragma unroll
  for (int it = 0; it < 2; ++it) {
    int q = tid + it * 128;              // [0,256): 8 pairs x 32 col-pairs
    int p = q >> 5, col = (q & 31) << 1;
    const float* g0p = Bb + (long long)(kc + 2 * p) * ldb + col0 + col;
    float2 g0 = *(const float2*)g0p;
    float2 g1 = *(const float2*)(g0p + ldb);
    *(float2*)&Bp[p][2 * col]     = make_float2(g0.x, g1.x);
    *(float2*)&Bp[p][2 * col + 2] = make_float2(g0.y, g1.y);
  }
}

// ---------------------------------------------------------------------------
// Generic batched fp32 WMMA GEMM:  C[b] = epilogue( A[b] * B[b] )
//   TA      : A stored as At[k][m] (k-row stride = lda); else A[m][k]
//   EPI     : 0 = none, 1 = +aux[col] (bias), 2 = *aux[row] (diag scale)
//   STORE_T : store C[col*ldc + row] (transposed) instead of C[row*ldc + col]
// 128 threads = 4 waves; wave w owns rows 32w..32w+31 (two 16-row groups),
// eight 16x16 f32 accumulators per wave; LDS double-buffered, 1 barrier/chunk.
// Requires M%128==0, N%64==0, K%16==0 (true for all launches below).
// ---------------------------------------------------------------------------
template<bool TA, int EPI, bool STORE_T>
__global__ __launch_bounds__(128)
void gemm_f32_wmma(const float* __restrict__ A, long long sA,
                   const float* __restrict__ Bm, long long sB,
                   float* __restrict__ C, long long sC,
                   const float* __restrict__ aux, long long sAux,
                   int M, int N, int K, int lda, int ldb, int ldc)
{
  (void)M; (void)N;
  __shared__ float As[2][BM][ALD];
  __shared__ float Bp[2][NPAIR][BPLD];

  const int b = blockIdx.z;
  const float* Ab = A  + (long long)b * sA;
  const float* Bb = Bm + (long long)b * sB;
  float*       Cb = C  + (long long)b * sC;

  const int tid  = threadIdx.x;
  const int wave = tid >> 5;
  const int lane = tid & 31;
  const int hl   = lane & 15;      // lane within 16-lane half
  const int hi   = lane >> 4;      // which half (0/1)

  const int row0 = blockIdx.x * BM;
  const int col0 = blockIdx.y * BN;

  v8f acc[2][4];
  #pragma unroll
  for (int g = 0; g < 2; ++g)
    #pragma unroll
    for (int c = 0; c < 4; ++c)
      #pragma unroll
      for (int v = 0; v < 8; ++v) acc[g][c][v] = 0.0f;

  // prologue stage
  if constexpr (!TA) stage_A_n(As[0], Ab, row0, lda, 0, tid);
  else               stage_A_t(As[0], Ab, row0, lda, 0, tid);
  stage_B_pairs(Bp[0], Bb, col0, ldb, 0, tid);

  int buf = 0;
  for (int kc = 0; kc < K; kc += BK) {
    __syncthreads();               // staged buf ready; other buf free
    if (kc + BK < K) {
      if constexpr (!TA) stage_A_n(As[buf ^ 1], Ab, row0, lda, kc + BK, tid);
      else               stage_A_t(As[buf ^ 1], Ab, row0, lda, kc + BK, tid);
      stage_B_pairs(Bp[buf ^ 1], Bb, col0, ldb, kc + BK, tid);
      if (kc + 2 * BK < K) {       // deepen pipeline: global_prefetch_b8
        if constexpr (!TA)
          __builtin_prefetch(Ab + (long long)(row0 + (tid >> 2)) * lda + kc + 2 * BK, 0, 3);
        else
          __builtin_prefetch(Ab + (long long)(kc + 2 * BK + (tid >> 4)) * lda + row0, 0, 3);
        __builtin_prefetch(Bb + (long long)(kc + 2 * BK + (tid >> 4)) * ldb + col0, 0, 3);
      }
    }
    // ---- 32 WMMAs per wave per chunk ----
    #pragma unroll
    for (int ks = 0; ks < BK; ks += 4) {
      const int ko = ks + (hi ? 2 : 0);
      const int p  = ko >> 1;
      v2f a0 = *(const v2f*)&As[buf][wave * 32 +      hl][ko];
      v2f a1 = *(const v2f*)&As[buf][wave * 32 + 16 + hl][ko];
      #pragma unroll
      for (int c = 0; c < 4; ++c) {
        v2f bf = *(const v2f*)&Bp[buf][p][(c * 16 + hl) << 1];
        acc[0][c] = __builtin_amdgcn_wmma_f32_16x16x4_f32(
            false, a0, false, bf, (short)0, acc[0][c], false, false);
        acc[1][c] = __builtin_amdgcn_wmma_f32_16x16x4_f32(
            false, a1, false, bf, (short)0, acc[1][c], false, false);
      }
    }
    buf ^= 1;
  }

  // ---- epilogue + store ----
  const int rbase = row0 + wave * 32;
  #pragma unroll
  for (int g = 0; g < 2; ++g) {
    #pragma unroll
    for (int c = 0; c < 4; ++c) {
      #pragma unroll
      for (int v = 0; v < 8; ++v) {
        int r  = rbase + g * 16 + v + (hi ? 8 : 0);
        int cl = col0 + c * 16 + hl;
        float val = acc[g][c][v];
        if constexpr (EPI == 1) val += aux[(long long)b * sAux + cl];
        if constexpr (EPI == 2) val *= aux[(long long)b * sAux + r];
        if constexpr (STORE_T) Cb[(long long)cl * ldc + r] = val;
        else                   Cb[(long long)r * ldc + cl] = val;
      }
    }
  }
}

// ---------------------------------------------------------------------------
// Fused attention: d[b,n] = softmax_m( zq[b,n,:] . zk[b,m,:] ) . ev[b,m]
// Flash-style: never materializes the [B,N,N] score matrix. Each workgroup
// owns 128 rows; loops over 64-column blocks, GEMM via WMMA (double-buffered
// LDS), then online (max, denom, ev-weighted numer) update with half-wave
// shuffle reductions (wave32: xor masks 1..8 stay inside each 16-lane half).
// ---------------------------------------------------------------------------
__global__ __launch_bounds__(128)
void attn_softmax_d(const float* __restrict__ zq,   // [B, Nn, Kd]
                    const float* __restrict__ zkT,  // [B, Kd, Nn]
                    const float* __restrict__ ev,   // [B, Nn]
                    float* __restrict__ dOut,       // [B, Nn]
                    int Nn, int Kd)
{
  __shared__ float As[2][BM][ALD];
  __shared__ float Bp[2][NPAIR][BPLD];

  const int b = blockIdx.z;
  const float* Ab  = zq  + (long long)b * Nn * Kd;
  const float* Bb  = zkT + (long long)b * Kd * Nn;
  const float* evb = ev  + (long long)b * Nn;

  const int tid  = threadIdx.x;
  const int wave = tid >> 5, lane = tid & 31, hl = lane & 15, hi = lane >> 4;
  const int row0 = blockIdx.x * BM;

  // running stats; slot [g][v] covers row (wave*32 + g*16 + v + 8*hi)
  float mrun[2][8], den[2][8], num[2][8];
  #pragma unroll
  for (int g = 0; g < 2; ++g)
    #pragma unroll
    for (int v = 0; v < 8; ++v) { mrun[g][v] = -INFINITY; den[g][v] = 0.f; num[g][v] = 0.f; }

  for (int m0 = 0; m0 < Nn; m0 += BN) {
    v8f acc[2][4];
    #pragma unroll
    for (int g = 0; g < 2; ++g)
      #pragma unroll
      for (int c = 0; c < 4; ++c)
        #pragma unroll
        for (int v = 0; v < 8; ++v) acc[g][c][v] = 0.0f;

    stage_A_n(As[0], Ab, row0, Kd, 0, tid);
    stage_B_pairs(Bp[0], Bb, m0, Nn, 0, tid);
    int buf = 0;
    for (int kc = 0; kc < Kd; kc += BK) {
      __syncthreads();
      if (kc + BK < Kd) {
        stage_A_n(As[buf ^ 1], Ab, row0, Kd, kc + BK, tid);
        stage_B_pairs(Bp[buf ^ 1], Bb, m0, Nn, kc + BK, tid);
        if (kc + 2 * BK < Kd) {
          __builtin_prefetch(Ab + (long long)(row0 + (tid >> 2)) * Kd + kc + 2 * BK, 0, 3);
          __builtin_prefetch(Bb + (long long)(kc + 2 * BK + (tid >> 4)) * Nn + m0, 0, 3);
        }
      }
      #pragma unroll
      for (int ks = 0; ks < BK; ks += 4) {
        const int ko = ks + (hi ? 2 : 0);
        const int p  = ko >> 1;
        v2f a0 = *(const v2f*)&As[buf][wave * 32 +      hl][ko];
        v2f a1 = *(const v2f*)&As[buf][wave * 32 + 16 + hl][ko];
        #pragma unroll
        for (int c = 0; c < 4; ++c) {
          v2f bf = *(const v2f*)&Bp[buf][p][(c * 16 + hl) << 1];
          acc[0][c] = __builtin_amdgcn_wmma_f32_16x16x4_f32(
              false, a0, false, bf, (short)0, acc[0][c], false, false);
          acc[1][c] = __builtin_amdgcn_wmma_f32_16x16x4_f32(
              false, a1, false, bf, (short)0, acc[1][c], false, false);
        }
      }
      buf ^= 1;
    }
    __syncthreads();   // done with both buffers before next m-block restages

    // online softmax update over this 64-column block
    float evc[4];
    #pragma unroll
    for (int c = 0; c < 4; ++c) evc[c] = evb[m0 + c * 16 + hl];

    #pragma unroll
    for (int g = 0; g < 2; ++g) {
      #pragma unroll
      for (int v = 0; v < 8; ++v) {
        float mx = fmaxf(fmaxf(acc[g][0][v], acc[g][1][v]),
                         fmaxf(acc[g][2][v], acc[g][3][v]));
        #pragma unroll
        for (int off = 1; off < 16; off <<= 1)
          mx = fmaxf(mx, __shfl_xor(mx, off, 32));
        float mnew = fmaxf(mrun[g][v], mx);
        float p0 = __expf(acc[g][0][v] - mnew), p1 = __expf(acc[g][1][v] - mnew);
        float p2 = __expf(acc[g][2][v] - mnew), p3 = __expf(acc[g][3][v] - mnew);
        float ds = p0 + p1 + p2 + p3;
        float ns = p0 * evc[0] + p1 * evc[1] + p2 * evc[2] + p3 * evc[3];
        #pragma unroll
        for (int off = 1; off < 16; off <<= 1) {
          ds += __shfl_xor(ds, off, 32);
          ns += __shfl_xor(ns, off, 32);
        }
        float sc = __expf(mrun[g][v] - mnew);  // 0 on first block (mrun=-inf)
        den[g][v] = den[g][v] * sc + ds;
        num[g][v] = num[g][v] * sc + ns;
        mrun[g][v] = mnew;
      }
    }
  }

  if (hl == 0) {   // lanes 0 and 16 write their rows
    #pragma unroll
    for (int g = 0; g < 2; ++g)
      #pragma unroll
      for (int v = 0; v < 8; ++v)
        dOut[(long long)b * Nn + row0 + wave * 32 + g * 16 + v + hi * 8] =
            num[g][v] / den[g][v];
  }
}

// ---------------------------------------------------------------------------
extern "C" void kernel_launch(void* const* d_in, const int* in_sizes, int n_in,
                              void* d_out, int out_size, void* d_ws, size_t ws_size,
                              hipStream_t stream)
{
  (void)in_sizes; (void)n_in; (void)out_size; (void)ws_size;
  const int Bb = 8, Nn = 2048, IN = 512, OUT = 512;

  const float* z  = (const float*)d_in[0];
  const float* ev = (const float*)d_in[1];
  const float* V  = (const float*)d_in[2];
  const float* W1 = (const float*)d_in[3];
  const float* b1 = (const float*)d_in[4];
  const float* W2 = (const float*)d_in[5];
  const float* b2 = (const float*)d_in[6];
  const float* Wo = (const float*)d_in[7];
  const float* bo = (const float*)d_in[8];
  float* out = (float*)d_out;

  // workspace: zq | zkT | d   (~64.1 MiB); t reuses zq, h reuses zkT
  float* zq   = (float*)d_ws;                   // [B, Nn, OUT]
  float* zkT  = zq  + (size_t)Bb * Nn * OUT;    // [B, OUT, Nn]
  float* dvec = zkT + (size_t)Bb * OUT * Nn;    // [B, Nn]
  float* t = zq;
  float* h = zkT;

  dim3 blk(128, 1, 1);

  // 1) zq = z @ W1 + b1
  gemm_f32_wmma<false, 1, false><<<dim3(Nn / BM, OUT / BN, Bb), blk, 0, stream>>>(
      z, (long long)Nn * IN, W1, 0, zq, (long long)Nn * OUT, b1, 0,
      Nn, OUT, IN, IN, OUT, OUT);

  // 2) zkT = (z @ W2 + b2)^T   (transposed store -> [OUT, Nn] per batch)
  gemm_f32_wmma<false, 1, true><<<dim3(Nn / BM, OUT / BN, Bb), blk, 0, stream>>>(
      z, (long long)Nn * IN, W2, 0, zkT, (long long)OUT * Nn, b2, 0,
      Nn, OUT, IN, IN, OUT, Nn);

  // 3) d = softmax(zq zk^T) . ev   (fused, never materializes [N,N])
  attn_softmax_d<<<dim3(Nn / BM, 1, Bb), blk, 0, stream>>>(zq, zkT, ev, dvec, Nn, OUT);

  // 4) t[m,i] = d[m] * sum_n V[n,m] z[n,i]   (A = V^T via TA, row-scale epilogue)
  gemm_f32_wmma<true, 2, false><<<dim3(Nn / BM, IN / BN, Bb), blk, 0, stream>>>(
      V, (long long)Nn * Nn, z, (long long)Nn * IN, t, (long long)Nn * IN, dvec, Nn,
      Nn, IN, Nn, Nn, IN, IN);

  // 5) h = V @ t
  gemm_f32_wmma<false, 0, false><<<dim3(Nn / BM, IN / BN, Bb), blk, 0, stream>>>(
      V, (long long)Nn * Nn, t, (long long)Nn * IN, h, (long long)Nn * IN, nullptr, 0,
      Nn, IN, Nn, Nn, IN, IN);

  // 6) out = h @ Wo + bo
  gemm_f32_wmma<false, 1, false><<<dim3(Nn / BM, OUT / BN, Bb), blk, 0, stream>>>(
      h, (long long)Nn * IN, Wo, 0, out, (long long)Nn * OUT, bo, 0,
      Nn, OUT, IN, IN, OUT, OUT);
}